// AdaFace_12738873000633
// MI455X (gfx1250) — compile-verified
//
#include <hip/hip_runtime.h>
#include <math.h>

// ---------------------------------------------------------------------------
// AdaFace margin kernel for MI455X (gfx1250).
//
// Bulk pass:  out = 64 * clamp(cosine, -cos(eps), cos(eps))   [exact identity
//             for cos(clip(acos(x), eps, pi-eps))], streamed through LDS with
//             the Tensor Data Mover (double-buffered tensor_load_to_lds),
//             NT stores on the write-once output stream.
// Fixup pass: exact acos/cos margin math on the 2048 label positions only.
// Stats pass: mean / unbiased std of clipped norms -> d_ws[0..1].
// ---------------------------------------------------------------------------

#define TILE 8192            // floats per TDM tile (32 KB); 2 buffers = 64 KB LDS
#define STREAM_GRID 1250

typedef unsigned int u32;
typedef u32 u32x4 __attribute__((ext_vector_type(4)));
typedef int i32x4 __attribute__((ext_vector_type(4)));
typedef int i32x8 __attribute__((ext_vector_type(8)));
typedef float f32x4 __attribute__((ext_vector_type(4)));   // native vector for NT store

// Clamp bounds: cos(1e-3) = 0.9999995; cos(pi - 1e-3) = -0.9999995
#define CHI 0.99999950f
#define CLO -0.99999950f
#define SCALE 64.0f
#define MARG 0.4f
#define HPAR 0.333f
#define EPS 1e-3f
#define PI_F 3.14159265358979f

// Issue one TDM load of TILE floats (tile index `tile` of `gbase`) into LDS at
// byte address `lds_addr`. Descriptor per CDNA5 ISA section 8 (D# groups).
__device__ __forceinline__ void tdm_load_tile(const float* gbase, int tile, u32 lds_addr) {
    unsigned long long ga =
        (unsigned long long)(size_t)gbase + (unsigned long long)tile * (TILE * 4ull);
    u32x4 g0;
    g0.x = 1u;                                            // count=1 (valid user D#)
    g0.y = lds_addr;                                      // lds_addr[31:0]
    g0.z = (u32)ga;                                       // global_addr[31:0]
    g0.w = (u32)((ga >> 32) & 0x1FFFFFFu) | (2u << 30);   // global_addr[56:32], type=2
    i32x8 g1;
    g1[0] = (int)(2u << 16);                              // data_size = 2 -> 4 bytes
    g1[1] = (int)(((u32)TILE & 0xFFFFu) << 16);           // tensor_dim0[15:0]
    g1[2] = (int)((((u32)TILE >> 16) & 0xFFFFu) | (1u << 16)); // tensor_dim0[31:16] | tensor_dim1[15:0]=1
    g1[3] = (int)((u32)TILE << 16);                       // tensor_dim1[31:16]=0 | tile_dim0
    g1[4] = 1;                                            // tile_dim1=1, tile_dim2=0
    g1[5] = TILE;                                         // tensor_dim0_stride[31:0]
    g1[6] = 0;                                            // stride0[47:32] | stride1[15:0]
    g1[7] = 0;                                            // stride1[47:16]
    i32x4 z4 = {0, 0, 0, 0};                              // groups 2/3 unused (2-D tensor)
    i32x8 z8 = {0, 0, 0, 0, 0, 0, 0, 0};                  // extra group (6-arg form), unused
    __builtin_amdgcn_tensor_load_to_lds(g0, g1, z4, z4, z8, 0);
}

__device__ __forceinline__ float clampscale(float x) {
    return fminf(fmaxf(x, CLO), CHI) * SCALE;
}

// --------------------------- bulk streaming pass ---------------------------
__global__ __launch_bounds__(256) void ada_stream(const float* __restrict__ in,
                                                  float* __restrict__ out,
                                                  int ntiles) {
    __shared__ float lds[2][TILE];
    const int tid = threadIdx.x;
    const int G = gridDim.x;
    int t = blockIdx.x;
    if (t >= ntiles) return;

    // Wave-uniform guard: SGPR compare -> scalar branch. TDM ops ignore EXEC,
    // so we must *branch around* them in waves 1..7, not predicate them.
    const bool dma = (__builtin_amdgcn_readfirstlane((u32)tid) == 0u);
    const u32 lbase0 = (u32)(size_t)&lds[0][0];
    const u32 lbase1 = (u32)(size_t)&lds[1][0];

    if (dma) tdm_load_tile(in, t, lbase0);

    int buf = 0;
    for (; t < ntiles; t += G, buf ^= 1) {
        const int tn = t + G;
        if (dma) {
            if (tn < ntiles) {
                tdm_load_tile(in, tn, buf ? lbase0 : lbase1);  // prefetch next tile
                __builtin_amdgcn_s_wait_tensorcnt((short)1);   // current tile landed
            } else {
                __builtin_amdgcn_s_wait_tensorcnt((short)0);
            }
        }
        __syncthreads();   // tile `t` visible in LDS to all 8 waves

        const f32x4* src = (const f32x4*)&lds[buf][0];
        f32x4* dst = (f32x4*)(out + (size_t)t * TILE);
#pragma unroll
        for (int k = 0; k < TILE / (4 * 256); ++k) {   // 8 x ds_load_b128 per thread
            f32x4 v = src[k * 256 + tid];
            v.x = clampscale(v.x);
            v.y = clampscale(v.y);
            v.z = clampscale(v.z);
            v.w = clampscale(v.w);
            __builtin_nontemporal_store(v, &dst[k * 256 + tid]);  // NT: don't rinse L2
        }
        __syncthreads();   // buffer free before wave 0 DMAs into it next iter
    }
}

// ------------------------------- tail pass ---------------------------------
__global__ __launch_bounds__(256) void ada_tail(const float* __restrict__ in,
                                                float* __restrict__ out,
                                                long long start, long long total) {
    long long i = start + (long long)blockIdx.x * blockDim.x + threadIdx.x;
    if (i < total) out[i] = clampscale(in[i]);
}

// ------------------------- norm statistics (d_ws) --------------------------
__global__ __launch_bounds__(256) void ada_stats(const float* __restrict__ norms,
                                                 int N, float* __restrict__ stats) {
    __shared__ float ss[256];
    __shared__ float sq[256];
    float s = 0.f, q = 0.f;
    for (int i = threadIdx.x; i < N; i += 256) {
        float v = fminf(fmaxf(norms[i], 1e-3f), 100.f);
        s += v;
        q += v * v;
    }
    ss[threadIdx.x] = s;
    sq[threadIdx.x] = q;
    __syncthreads();
    for (int off = 128; off > 0; off >>= 1) {
        if (threadIdx.x < off) {
            ss[threadIdx.x] += ss[threadIdx.x + off];
            sq[threadIdx.x] += sq[threadIdx.x + off];
        }
        __syncthreads();
    }
    if (threadIdx.x == 0) {
        float sum = ss[0], sumsq = sq[0];
        float mean = sum / (float)N;
        float var = (sumsq - sum * sum / (float)N) / (float)(N - 1);  // unbiased
        stats[0] = mean;
        stats[1] = sqrtf(fmaxf(var, 0.f));
    }
}

// --------------------- exact margin math at label cells --------------------
__global__ __launch_bounds__(256) void ada_fixup(const float* __restrict__ cosine,
                                                 const float* __restrict__ norms,
                                                 const int* __restrict__ label,
                                                 const float* __restrict__ stats,
                                                 float* __restrict__ out,
                                                 int N, int C) {
    int i = blockIdx.x * blockDim.x + threadIdx.x;
    if (i >= N) return;
    int lab = label[i];
    if (lab < 0) return;  // invalid row: no margin applied
    float sn = fminf(fmaxf(norms[i], 1e-3f), 100.f);
    float ms = (sn - stats[0]) / (stats[1] + EPS) * HPAR;
    ms = fminf(fmaxf(ms, -1.f), 1.f);
    size_t idx = (size_t)i * (size_t)C + (size_t)lab;
    float x = cosine[idx];
    float theta = acosf(x) - MARG * ms;                      // + g_angular
    theta = fminf(fmaxf(theta, EPS), PI_F - EPS);            // row clip
    float cm = cosf(theta) - (MARG + MARG * ms);             // - g_add
    out[idx] = cm * SCALE;
}

// ------------------------------- launcher ----------------------------------
extern "C" void kernel_launch(void* const* d_in, const int* in_sizes, int n_in,
                              void* d_out, int out_size, void* d_ws, size_t ws_size,
                              hipStream_t stream) {
    const float* cosine = (const float*)d_in[0];
    const float* norms  = (const float*)d_in[1];
    const int*   label  = (const int*)d_in[2];
    float* out   = (float*)d_out;
    float* stats = (float*)d_ws;

    const int N = in_sizes[1];                       // 2048 (norms is N x 1)
    const long long total = (long long)in_sizes[0];  // N * C elements
    const int C = (int)(total / (long long)N);
    const int ntiles = (int)(total / TILE);
    const long long rem_start = (long long)ntiles * TILE;

    ada_stats<<<1, 256, 0, stream>>>(norms, N, stats);

    if (ntiles > 0) {
        int grid = ntiles < STREAM_GRID ? ntiles : STREAM_GRID;
        ada_stream<<<grid, 256, 0, stream>>>(cosine, out, ntiles);
    }
    if (rem_start < total) {
        long long rem = total - rem_start;
        int blocks = (int)((rem + 255) / 256);
        ada_tail<<<blocks, 256, 0, stream>>>(cosine, out, rem_start, total);
    }
    ada_fixup<<<(N + 255) / 256, 256, 0, stream>>>(cosine, norms, label, stats, out, N, C);
}